// GraphConv2d_32607391711980
// MI455X (gfx1250) — compile-verified
//
#include <hip/hip_runtime.h>
#include <hip/hip_bf16.h>
#include <math.h>

typedef __attribute__((ext_vector_type(2))) float v2f;
typedef __attribute__((ext_vector_type(4))) float v4f;
typedef __attribute__((ext_vector_type(8))) float v8f;

#define BDIM   2
#define CDIM   128
#define NDIM   10000
#define KDIM   16
#define NTILES 625   /* NDIM / 16 */

// ---------------------------------------------------------------------------
// Kernel 1: whT[b][n][o] = sum_c W[o][c] * x[b][c][n]   (node-major output)
//           si[b][n] = sum_o a[o]      * wh[b][o][n]
//           sj[b][n] = sum_o a[C + o]  * wh[b][o][n]
// 256 threads = 8 waves; each wave owns one 16-node tile and iterates the
// 8 output-channel tiles, doing 32x V_WMMA_F32_16X16X4_F32 per tile.
// W (64KB) staged in LDS, shared by all waves in the workgroup.
// ---------------------------------------------------------------------------
__global__ __launch_bounds__(256)
void gat_gemm_kernel(const float* __restrict__ x, const float* __restrict__ W,
                     const float* __restrict__ a,
                     float* __restrict__ whT, float* __restrict__ si,
                     float* __restrict__ sj)
{
    __shared__ float Wl[CDIM * CDIM];   // 64 KB
    __shared__ float al[2 * CDIM];      // 1 KB

    const int tid = threadIdx.x;
    {   // cooperative load of W and a into LDS
        const float4* Wg4 = (const float4*)W;
        float4*       Wl4 = (float4*)Wl;
        #pragma unroll 4
        for (int i = tid; i < CDIM * CDIM / 4; i += 256) Wl4[i] = Wg4[i];
        al[tid] = a[tid];   // exactly 256 threads / 256 values
    }
    __syncthreads();

    const int wave = tid >> 5;
    const int lane = tid & 31;
    const int tile = blockIdx.x * 8 + wave;
    if (tile >= NTILES) return;

    const int b  = blockIdx.y;
    const int n0 = tile * 16;
    const int ln = lane & 15;          // node column within tile / A row M
    const int kb = (lane >> 4) * 2;    // K offset for this lane-half (0 or 2)

    // Preload x B-fragments for all 32 K-steps (B layout: K = kb+{0,1}, N = ln)
    v2f xb[32];
    const float* xcol = x + (size_t)b * CDIM * NDIM + n0 + ln;
    #pragma unroll
    for (int ks = 0; ks < 32; ++ks) {
        const int c0 = ks * 4 + kb;
        xb[ks].x = xcol[(size_t)(c0 + 0) * NDIM];
        xb[ks].y = xcol[(size_t)(c0 + 1) * NDIM];
    }

    float sip = 0.f, sjp = 0.f;
    float* outrow = whT + ((size_t)b * NDIM + n0 + ln) * CDIM;

    #pragma unroll 1
    for (int ot = 0; ot < 8; ++ot) {
        const int o0 = ot * 16;
        v8f acc = {};
        // A fragment: W[m][k], W[m][k+1] with m = o0+ln, k = 4*ks + kb (LDS)
        const float* Wrow = Wl + (o0 + ln) * CDIM + kb;
        #pragma unroll
        for (int ks = 0; ks < 32; ++ks) {
            v2f av = *(const v2f*)(Wrow + ks * 4);
            acc = __builtin_amdgcn_wmma_f32_16x16x4_f32(
                      false, av, false, xb[ks], (short)0, acc, false, false);
        }
        // D layout: lane holds M = o0 + 8*(lane>>4) + v, N = ln
        const int oset = o0 + 8 * (lane >> 4);
        v4f lo = { acc[0], acc[1], acc[2], acc[3] };
        v4f hi = { acc[4], acc[5], acc[6], acc[7] };
        *(v4f*)(outrow + oset)     = lo;
        *(v4f*)(outrow + oset + 4) = hi;
        #pragma unroll
        for (int v = 0; v < 8; ++v) {
            sip += al[oset + v]        * acc[v];
            sjp += al[CDIM + oset + v] * acc[v];
        }
    }
    // combine the two lane-halves that cover the same node
    sip += __shfl_xor(sip, 16);
    sjp += __shfl_xor(sjp, 16);
    if (lane < 16) {
        si[(size_t)b * NDIM + n0 + lane] = sip;
        sj[(size_t)b * NDIM + n0 + lane] = sjp;
    }
}

// ---------------------------------------------------------------------------
// Kernel 2: per node n — logits via scalar gathers of si/sj, leaky-relu,
// softmax over K=16 (wave shuffles), then h[:,n] = sum_k A_k * whT[j_k,:].
// 512 threads = 16 waves, one node per wave; LDS transpose for coalesced out.
// ---------------------------------------------------------------------------
__global__ __launch_bounds__(512)
void gat_aggr_kernel(const int* __restrict__ ei, const float* __restrict__ whT,
                     const float* __restrict__ si, const float* __restrict__ sj,
                     float* __restrict__ out)
{
    __shared__ float hbuf[16][CDIM + 4];

    const int tid  = threadIdx.x;
    const int wave = tid >> 5;
    const int lane = tid & 31;
    const int b    = blockIdx.y;
    const int n0   = blockIdx.x * 16;
    const int n    = n0 + wave;

    // edge_index: [2, B, N, K]; row 0 = j (source), row 1 = i (dest)
    const size_t eb = ((size_t)b * NDIM + n) * KDIM;
    int   jj = 0;
    float e  = -3.0e30f;
    if (lane < 16) {
        jj = ei[eb + lane];
        const int ii = ei[(size_t)BDIM * NDIM * KDIM + eb + lane];
        const float ev = si[(size_t)b * NDIM + ii] + sj[(size_t)b * NDIM + jj];
        e = (ev > 0.f) ? ev : 0.2f * ev;   // leaky_relu(0.2)
    }
    // softmax over lanes 0..15 (xor masks stay within the half)
    float m = e;
    m = fmaxf(m, __shfl_xor(m, 8));
    m = fmaxf(m, __shfl_xor(m, 4));
    m = fmaxf(m, __shfl_xor(m, 2));
    m = fmaxf(m, __shfl_xor(m, 1));
    float p = (lane < 16) ? __expf(e - m) : 0.f;
    float s = p;
    s += __shfl_xor(s, 8);
    s += __shfl_xor(s, 4);
    s += __shfl_xor(s, 2);
    s += __shfl_xor(s, 1);
    const float A = p / s;

    // aggregation: each lane owns 4 channels, loop over 16 neighbors
    v4f h = { 0.f, 0.f, 0.f, 0.f };
    const float* wb = whT + (size_t)b * NDIM * CDIM;
    #pragma unroll
    for (int k = 0; k < 16; ++k) {
        const float ak = __shfl(A, k);
        const int   jk = __shfl(jj, k);
        const v4f w4 = *(const v4f*)(wb + (size_t)jk * CDIM + 4 * lane);
        h += ak * w4;                      // wave reads 512B contiguous
    }
    *(v4f*)(&hbuf[wave][4 * lane]) = h;
    __syncthreads();

    // transpose: thread t -> channel c = t>>2, node group q = t&3
    const int c = tid >> 2;
    const int q = tid & 3;
    v4f o4 = { hbuf[4 * q + 0][c], hbuf[4 * q + 1][c],
               hbuf[4 * q + 2][c], hbuf[4 * q + 3][c] };
    *(v4f*)(out + ((size_t)b * CDIM + c) * NDIM + n0 + 4 * q) = o4;
}

// ---------------------------------------------------------------------------
extern "C" void kernel_launch(void* const* d_in, const int* in_sizes, int n_in,
                              void* d_out, int out_size, void* d_ws, size_t ws_size,
                              hipStream_t stream)
{
    const float* x  = (const float*)d_in[0];   // [B,C,N,1]
    const int*   ei = (const int*)  d_in[1];   // [2,B,N,K]
    const float* W  = (const float*)d_in[2];   // [C,C]
    const float* a  = (const float*)d_in[3];   // [2C]
    float* out = (float*)d_out;                // [B,C,N,1]

    float* whT = (float*)d_ws;                         // B*N*C floats (10.24 MB)
    float* si  = whT + (size_t)BDIM * NDIM * CDIM;     // B*N floats
    float* sj  = si  + (size_t)BDIM * NDIM;            // B*N floats

    dim3 g1((NTILES + 7) / 8, BDIM);
    gat_gemm_kernel<<<g1, 256, 0, stream>>>(x, W, a, whT, si, sj);

    dim3 g2(NTILES, BDIM);
    gat_aggr_kernel<<<g2, 512, 0, stream>>>(ei, whT, si, sj, out);
}